// QuantizedLinear_50414326120584
// MI455X (gfx1250) — compile-verified
//
#include <hip/hip_runtime.h>

// ---------------------------------------------------------------------------
// QuantizedLinear for MI455X (gfx1250):
//   out[b,s,o] = sum_i x[b,s,i] * Wcat[o, inv_col_perm[i]] + bias[o]
// (1) prep kernel builds the dequantized+permuted weight in K-PAIR-
//     INTERLEAVED layout Wp[k/2][n] = (W[k][n], W[k+1][n]) into d_ws, so the
//     GEMM's B operand is a single contiguous ds_load_b64 per WMMA (no
//     register repacking);
// (2) fp32 WMMA GEMM (memory-bound at ~14us HBM floor, so full-precision
//     V_WMMA_F32_16X16X4_F32 is free vs HBM), 128x128 block tiles,
//     double-buffered LDS fed by CDNA5 async global->LDS copies (ASYNCcnt).
// ---------------------------------------------------------------------------

#define IN_F      4096
#define OUT_F     4096
#define OUTLIER   128
#define Q_IN      (IN_F - OUTLIER)   // 3968
#define BATCH_M   (4 * 2048)         // 8192 rows

typedef __attribute__((ext_vector_type(2))) float v2f;
typedef __attribute__((ext_vector_type(4))) float f4;
typedef __attribute__((ext_vector_type(8))) float v8f;
typedef int v4i __attribute__((vector_size(16)));   // matches builtin param type

#if defined(__gfx1250__) && __has_builtin(__builtin_amdgcn_global_load_async_to_lds_b128)
#define HAVE_ASYNC 1
#else
#define HAVE_ASYNC 0
#endif

// 16-byte global -> LDS copy. Async path bypasses VGPRs and is tracked by
// ASYNCcnt; fallback is a plain synchronous copy through registers.
__device__ __forceinline__ void copy16(const float* g, float* l) {
#if HAVE_ASYNC
    __builtin_amdgcn_global_load_async_to_lds_b128((v4i*)g, (v4i*)l, 0, 0);
#else
    *(f4*)l = *(const f4*)g;
#endif
}

template <int N>
__device__ __forceinline__ void wait_async_le() {
#if HAVE_ASYNC
#if __has_builtin(__builtin_amdgcn_s_wait_asynccnt)
    __builtin_amdgcn_s_wait_asynccnt(N);
#else
    asm volatile("s_wait_asynccnt %0" ::"n"(N));
#endif
#endif
}

// ---------------------------------------------------------------------------
// Kernel 1: dequantize + permute, emitting K-pair-interleaved layout:
//   Wp[((i>>1)*OUT_F + o)*2 + (i&1)] = Wcat[o, inv_col_perm[i]]
// 64(i) x 64(o) tile staged through LDS; the (even,odd) i-pair is fused into
// one float2 store so global writes stay fully coalesced.
// ---------------------------------------------------------------------------
__global__ __launch_bounds__(256) void ql_prep_weight(
    const float* __restrict__ q_w,      // [OUT_F][Q_IN]
    const float* __restrict__ fp_w,     // [OUT_F][OUTLIER]
    const float* __restrict__ alpha,    // [OUT_F]
    const int*   __restrict__ inv_perm, // [IN_F]
    float*       __restrict__ Wp)       // [IN_F/2][OUT_F][2]
{
    __shared__ int   jmap[64];
    __shared__ float tile[64][65];

    const int t  = threadIdx.x;
    const int i0 = blockIdx.x * 64;
    const int o0 = blockIdx.y * 64;

    if (t < 64) jmap[t] = inv_perm[i0 + t];
    __syncthreads();

    const int il = t & 63;
    const int ty = t >> 6;              // 0..3
    const int j  = jmap[il];

#pragma unroll
    for (int oo = 0; oo < 16; ++oo) {
        const int o = o0 + ty * 16 + oo;
        float v;
        if (j < Q_IN) v = q_w[(size_t)o * Q_IN + j] * alpha[o];
        else          v = fp_w[(size_t)o * OUTLIER + (j - Q_IN)];
        tile[il][ty * 16 + oo] = v;
    }
    __syncthreads();

    const int ol = t & 63;
#pragma unroll
    for (int ii = 0; ii < 8; ++ii) {
        const int il2 = ty * 16 + 2 * ii;        // even i in pair
        v2f w;
        w.x = tile[il2 + 0][ol];
        w.y = tile[il2 + 1][ol];
        const size_t idx =
            ((size_t)((i0 + il2) >> 1) * OUT_F + (o0 + ol)) * 2;
        *(v2f*)&Wp[idx] = w;                      // 8B-aligned pair store
    }
}

// ---------------------------------------------------------------------------
// Kernel 2: fp32 WMMA GEMM.  C[M,N] = X[M,K] * W[K,N] + bias.
//   256 threads = 8 waves. Block tile 128(M) x 128(N), K-step 16.
//   Wave (wm,wn) owns a 32(M) x 64(N) strip -> acc[2][4] of v8f.
//   Double-buffered LDS; next tile streamed via async global->LDS while the
//   current one feeds the WMMAs.  Each wave issues NPER async copies per
//   tile; waiting ASYNCcnt<=NPER guarantees the current tile landed.
//   Per ISA wave32 layouts: A lane = row M=lane%16, K pair 2*(lane/16);
//   B lane = col N=lane%16, same K pair; C VGPR v -> M = v + 8*(lane/16).
//   LDS strides: Xs 20 floats (80B), Ws 260 floats (1040B) -- both 16B
//   multiples; A and B operands are single contiguous b64 LDS loads.
// ---------------------------------------------------------------------------
#define TILE_M 128
#define TILE_N 128
#define TILE_K 16
#define NPER   4     // async copy instructions per wave per K-tile

__global__ __launch_bounds__(256) void ql_wmma_gemm(
    const float* __restrict__ X,    // [M][K]
    const float* __restrict__ Wp,   // [K/2][N][2]  pair-interleaved
    const float* __restrict__ bias, // [N]
    float*       __restrict__ out,  // [M][N]
    int M, int N, int K)
{
    __shared__ float Xs[2][TILE_M][TILE_K + 4];        // 2 x 128 x 20 floats
    __shared__ float Ws[2][TILE_K / 2][2 * TILE_N + 4];// 2 x 8 x 260 floats

    const int tid  = threadIdx.x;
    const int lane = tid & 31;
    const int wid  = tid >> 5;          // 0..7
    const int lrow = lane & 15;
    const int lhi  = lane >> 4;
    const int wm   = (wid & 3) * 32;    // wave M offset in block tile
    const int wn   = (wid >> 2) * 64;   // wave N offset in block tile

    const int m0 = blockIdx.y * TILE_M;
    const int n0 = blockIdx.x * TILE_N;

    v8f acc[2][4] = {};

    auto load_tiles = [&](int bsel, int k0) {
        // X tile: 128 rows x 16 cols = 512 x float4
#pragma unroll
        for (int r = 0; r < 2; ++r) {
            const int e   = tid + r * 256;
            const int row = e >> 2;             // 0..127
            const int c4  = (e & 3) << 2;       // 0,4,8,12
            copy16(X + (size_t)(m0 + row) * K + (k0 + c4), &Xs[bsel][row][c4]);
        }
        // W tile: 8 pair-rows x 256 floats = 512 x float4
#pragma unroll
        for (int r = 0; r < 2; ++r) {
            const int e   = tid + r * 256;
            const int row = e >> 6;             // 0..7   (k-pair row)
            const int c4  = (e & 63) << 2;      // 0..252 (floats within row)
            const size_t g = (size_t)((k0 >> 1) + row) * (2 * (size_t)N)
                           + 2 * (size_t)n0 + c4;
            copy16(Wp + g, &Ws[bsel][row][c4]);
        }
    };

    load_tiles(0, 0);                    // prologue: tile 0 -> buffer 0
    int buf = 0;
    for (int k0 = 0; k0 < K; k0 += TILE_K) {
        if (k0 + TILE_K < K) {
            load_tiles(buf ^ 1, k0 + TILE_K);  // stream next tile
            wait_async_le<NPER>();             // current tile done, next in flight
        } else {
            wait_async_le<0>();                // last tile: drain
        }
        __syncthreads();

#pragma unroll
        for (int kk = 0; kk < TILE_K; kk += 4) {
            v2f a[2];
#pragma unroll
            for (int i = 0; i < 2; ++i)
                a[i] = *(const v2f*)&Xs[buf][wm + i * 16 + lrow][kk + 2 * lhi];
            v2f b[4];
#pragma unroll
            for (int j = 0; j < 4; ++j)
                b[j] = *(const v2f*)
                    &Ws[buf][(kk >> 1) + lhi][2 * (wn + j * 16 + lrow)];
#pragma unroll
            for (int i = 0; i < 2; ++i)
#pragma unroll
                for (int j = 0; j < 4; ++j)
                    acc[i][j] = __builtin_amdgcn_wmma_f32_16x16x4_f32(
                        false, a[i], false, b[j], (short)0, acc[i][j],
                        false, false);
        }
        __syncthreads();
        buf ^= 1;
    }

    // ---- bias + writeback ----
#pragma unroll
    for (int j = 0; j < 4; ++j) {
        const int n  = n0 + wn + j * 16 + lrow;
        const float bv = bias[n];
#pragma unroll
        for (int i = 0; i < 2; ++i) {
#pragma unroll
            for (int v = 0; v < 8; ++v) {
                const int m = m0 + wm + i * 16 + v + 8 * lhi;
                out[(size_t)m * N + n] = acc[i][j][v] + bv;
            }
        }
    }
}

// ---------------------------------------------------------------------------
// Host-side launcher
// ---------------------------------------------------------------------------
extern "C" void kernel_launch(void* const* d_in, const int* in_sizes, int n_in,
                              void* d_out, int out_size, void* d_ws, size_t ws_size,
                              hipStream_t stream) {
    const float* x     = (const float*)d_in[0];  // [B,S,IN_F]
    const float* q_w   = (const float*)d_in[1];  // [OUT_F,Q_IN]
    const float* fp_w  = (const float*)d_in[2];  // [OUT_F,OUTLIER]
    const float* alpha = (const float*)d_in[3];  // [OUT_F,1]
    const float* bias  = (const float*)d_in[4];  // [OUT_F]
    const int*   inv   = (const int*)  d_in[5];  // [IN_F]
    float* out = (float*)d_out;                  // [B,S,OUT_F]
    float* Wp  = (float*)d_ws;                   // [IN_F/2][OUT_F][2] = 64 MB

    dim3 gPrep(IN_F / 64, OUT_F / 64);
    ql_prep_weight<<<gPrep, 256, 0, stream>>>(q_w, fp_w, alpha, inv, Wp);

    dim3 gGemm(OUT_F / TILE_N, BATCH_M / TILE_M);
    ql_wmma_gemm<<<gGemm, 256, 0, stream>>>(x, Wp, bias, out,
                                            BATCH_M, OUT_F, IN_F);
}